// BaseNet_12601434046511
// MI455X (gfx1250) — compile-verified
//
#include <hip/hip_runtime.h>
#include <math.h>

// ---------------------------------------------------------------------------
// GCN (3x GCNConv width 32) + first-node readout + MLP + log_softmax
// Sizes fixed by the reference: N=100000, E=1600000, D_IN=128, LAT=32, G=1000
// ---------------------------------------------------------------------------

typedef __attribute__((ext_vector_type(2))) float v2f;
typedef __attribute__((ext_vector_type(8))) float v8f;

#define GCN_N     100000
#define GCN_E     1600000
#define GCN_DIN   128
#define GCN_LAT   32
#define GCN_G     1000
#define GCN_HID   128

// ---------------- degree / normalization ----------------

__global__ void fill_ones(float* __restrict__ deg, int n) {
    int i = blockIdx.x * blockDim.x + threadIdx.x;
    if (i < n) deg[i] = 1.0f;
}

__global__ void deg_accum(const int* __restrict__ ei, float* __restrict__ deg, int e) {
    int i = blockIdx.x * blockDim.x + threadIdx.x;
    if (i < e) atomicAdd(&deg[ei[e + i]], 1.0f);   // dst = ei[1*E + i]
}

__global__ void rsqrt_inplace(float* __restrict__ d, int n) {
    int i = blockIdx.x * blockDim.x + threadIdx.x;
    if (i < n) d[i] = rsqrtf(d[i]);
}

// ---------------- WMMA fp32 transform: H[M,32] = A[M,K] @ W[K,32] ----------
// One wave computes a 16x32 tile: two 16x16 f32 accumulators, K stepped by 4
// via V_WMMA_F32_16X16X4_F32 (exact fp32 -> matches reference numerics).
//
// A frag (16x4 f32, 2 VGPRs): lane m=lane&15; lanes<16 carry K=k0+0..1,
// lanes>=16 carry K=k0+2..3  -> per-lane contiguous float2 load.
// B frag (4x16): same K striping across lane halves, N = lane&15.
// C (16x16, 8 VGPRs): vgpr v -> row v (lanes 0-15) / row v+8 (lanes 16-31).

template<int K>
__global__ void gemm_wmma_16x32(const float* __restrict__ A,
                                const float* __restrict__ W,
                                float* __restrict__ H, int M) {
    const int lane = threadIdx.x & 31;
    const int wave = (blockIdx.x * blockDim.x + threadIdx.x) >> 5;
    const int row0 = wave * 16;
    if (row0 >= M) return;                  // wave-uniform: EXEC all-1s inside

    const int hi = lane >> 4;               // 0 or 1 (K sub-pair)
    const int n  = lane & 15;
    const int m  = row0 + n;                // A row for this lane

    // prefetch next row-tile of A (speculative; OOB silently dropped)
    __builtin_prefetch(A + (size_t)(m + 16) * K, 0, 1);

    v8f c0 = {}; v8f c1 = {};
    #pragma unroll
    for (int k0 = 0; k0 < K; k0 += 4) {
        const int ka = k0 + 2 * hi;
        v2f a = *(const v2f*)(A + (size_t)m * K + ka);
        v2f b0, b1;
        b0.x = W[ka * 32 + n];        b0.y = W[(ka + 1) * 32 + n];
        b1.x = W[ka * 32 + 16 + n];   b1.y = W[(ka + 1) * 32 + 16 + n];
        c0 = __builtin_amdgcn_wmma_f32_16x16x4_f32(false, a, false, b0,
                                                   (short)0, c0, false, false);
        c1 = __builtin_amdgcn_wmma_f32_16x16x4_f32(false, a, false, b1,
                                                   (short)0, c1, false, false);
    }

    const int base_row = row0 + 8 * hi;     // rows v / v+8 split by lane half
    #pragma unroll
    for (int v = 0; v < 8; ++v) {
        H[(size_t)(base_row + v) * 32 + n]      = c0[v];
        H[(size_t)(base_row + v) * 32 + 16 + n] = c1[v];
    }
}

// ---------------- aggregation ----------------

// agg[i][d] = h[i][d] * (dis[i]*dis[i]) + b[d]
__global__ void init_agg(const float* __restrict__ h, const float* __restrict__ dis,
                         const float* __restrict__ b, float* __restrict__ agg, int n) {
    int i = blockIdx.x * blockDim.x + threadIdx.x;
    if (i < n * 32) {
        int node = i >> 5, d = i & 31;
        float s = dis[node];
        agg[i] = fmaf(h[i], s * s, b[d]);
    }
}

// one wave per edge, lane = feature dim (coalesced 128B gather + scatter)
__global__ void edge_agg(const int* __restrict__ ei, const float* __restrict__ dis,
                         const float* __restrict__ h, float* __restrict__ agg, int e) {
    int edge = blockIdx.x * (blockDim.x >> 5) + (threadIdx.x >> 5);
    if (edge >= e) return;
    int t   = threadIdx.x & 31;
    int src = ei[edge];
    int dst = ei[e + edge];
    float coef = dis[src] * dis[dst];
    float val  = h[(size_t)src * 32 + t] * coef;
    atomicAdd(&agg[(size_t)dst * 32 + t], val);
}

__global__ void tanh_k(const float* __restrict__ in, float* __restrict__ out, int n) {
    int i = blockIdx.x * blockDim.x + threadIdx.x;
    if (i < n) out[i] = tanhf(in[i]);
}

// ---------------- readout: first node per graph -> MLP -> log_softmax -----

__global__ void readout(const int* __restrict__ batch,
                        const float* __restrict__ h1, const float* __restrict__ h2,
                        const float* __restrict__ h3,
                        const float* __restrict__ l1w, const float* __restrict__ l1b,
                        const float* __restrict__ l2w, const float* __restrict__ l2b,
                        float* __restrict__ out, int n) {
    __shared__ float sgf[96];
    __shared__ float sh[GCN_HID];
    __shared__ int sidx;
    const int g = blockIdx.x;
    const int t = threadIdx.x;

    if (t == 0) {                            // searchsorted(batch, g)
        int lo = 0, hh = n;
        while (lo < hh) { int mid = (lo + hh) >> 1;
                          if (batch[mid] < g) lo = mid + 1; else hh = mid; }
        sidx = lo;
    }
    __syncthreads();
    const int idx = sidx;

    if (t < 96) {                            // cat = [h1 | h2 | h3] at idx
        float v;
        if (t < 32)      v = h1[(size_t)idx * 32 + t];
        else if (t < 64) v = h2[(size_t)idx * 32 + (t - 32)];
        else             v = h3[(size_t)idx * 32 + (t - 64)];
        sgf[t] = v;
    }
    __syncthreads();

    float acc = l1b[t];                      // hidden[t] = relu(g @ lin1_w + b)
    #pragma unroll 4
    for (int j = 0; j < 96; ++j) acc = fmaf(sgf[j], l1w[j * GCN_HID + t], acc);
    sh[t] = fmaxf(acc, 0.0f);
    __syncthreads();

    if (t == 0) {                            // logits + stable log_softmax
        float z0 = l2b[0], z1 = l2b[1];
        for (int k = 0; k < GCN_HID; ++k) {
            z0 = fmaf(sh[k], l2w[k * 2 + 0], z0);
            z1 = fmaf(sh[k], l2w[k * 2 + 1], z1);
        }
        float mx  = fmaxf(z0, z1);
        float lse = mx + logf(expf(z0 - mx) + expf(z1 - mx));
        out[(size_t)g * 2 + 0] = z0 - lse;
        out[(size_t)g * 2 + 1] = z1 - lse;
    }
}

// ---------------- driver ----------------

extern "C" void kernel_launch(void* const* d_in, const int* in_sizes, int n_in,
                              void* d_out, int out_size, void* d_ws, size_t ws_size,
                              hipStream_t stream) {
    const float* x      = (const float*)d_in[0];
    const int*   ei     = (const int*)  d_in[1];   // [2,E]
    const int*   batch  = (const int*)  d_in[2];   // [N]
    /* d_in[3] = num_graphs (device scalar) — value fixed at 1000 */
    const float* W1 = (const float*)d_in[4];  const float* b1 = (const float*)d_in[5];
    const float* W2 = (const float*)d_in[6];  const float* b2 = (const float*)d_in[7];
    const float* W3 = (const float*)d_in[8];  const float* b3 = (const float*)d_in[9];
    const float* l1w = (const float*)d_in[10]; const float* l1b = (const float*)d_in[11];
    const float* l2w = (const float*)d_in[12]; const float* l2b = (const float*)d_in[13];
    float* out = (float*)d_out;

    const int N = in_sizes[2];          // 100000
    const int E = in_sizes[1] / 2;      // 1600000

    // workspace layout (floats): dis | h_lin | agg | h1 | h2 | h3
    float* dis  = (float*)d_ws;
    float* hlin = dis  + N;
    float* agg  = hlin + (size_t)N * 32;
    float* h1   = agg  + (size_t)N * 32;
    float* h2   = h1   + (size_t)N * 32;
    float* h3   = h2   + (size_t)N * 32;

    const int TB = 256;
    const int nBlkN  = (N + TB - 1) / TB;
    const int nBlkE  = (E + TB - 1) / TB;
    const int nBlkNF = (N * 32 + TB - 1) / TB;
    const int gemmBlk = ((N / 16) + 7) / 8;        // 8 waves (16-row tiles) / block
    const int edgeBlk = (E + 7) / 8;               // 8 edges (waves) / block

    // deg^{-1/2} with self-loops
    fill_ones    <<<nBlkN, TB, 0, stream>>>(dis, N);
    deg_accum    <<<nBlkE, TB, 0, stream>>>(ei, dis, E);
    rsqrt_inplace<<<nBlkN, TB, 0, stream>>>(dis, N);

    // layer 1: K = 128
    gemm_wmma_16x32<GCN_DIN><<<gemmBlk, TB, 0, stream>>>(x, W1, hlin, N);
    init_agg<<<nBlkNF, TB, 0, stream>>>(hlin, dis, b1, agg, N);
    edge_agg<<<edgeBlk, TB, 0, stream>>>(ei, dis, hlin, agg, E);
    tanh_k  <<<nBlkNF, TB, 0, stream>>>(agg, h1, N * 32);

    // layer 2: K = 32
    gemm_wmma_16x32<GCN_LAT><<<gemmBlk, TB, 0, stream>>>(h1, W2, hlin, N);
    init_agg<<<nBlkNF, TB, 0, stream>>>(hlin, dis, b2, agg, N);
    edge_agg<<<edgeBlk, TB, 0, stream>>>(ei, dis, hlin, agg, E);
    tanh_k  <<<nBlkNF, TB, 0, stream>>>(agg, h2, N * 32);

    // layer 3: K = 32
    gemm_wmma_16x32<GCN_LAT><<<gemmBlk, TB, 0, stream>>>(h2, W3, hlin, N);
    init_agg<<<nBlkNF, TB, 0, stream>>>(hlin, dis, b3, agg, N);
    edge_agg<<<edgeBlk, TB, 0, stream>>>(ei, dis, hlin, agg, E);
    tanh_k  <<<nBlkNF, TB, 0, stream>>>(agg, h3, N * 32);

    // readout + MLP + log_softmax
    readout<<<GCN_G, GCN_HID, 0, stream>>>(batch, h1, h2, h3,
                                           l1w, l1b, l2w, l2b, out, N);
    (void)n_in; (void)out_size; (void)ws_size; (void)d_out;
}